// RoIAlign_42133629173787
// MI455X (gfx1250) — compile-verified
//
#include <hip/hip_runtime.h>
#include <hip/hip_bf16.h>
#include <stdint.h>

// Problem constants (fixed by the reference setup).
constexpr int N_ = 8, C_ = 256, H_ = 100, W_ = 152;
constexpr int PH_ = 7, PW_ = 7;
constexpr float SCALE_ = 0.25f;
constexpr int HW_ = H_ * W_;          // 15200 = 95 * 160 = 475 * 32

// TDM transpose tile: 32 channels x 160 pixels, LDS pitch 165 (hw padded).
constexpr int TW_ = 160;              // tile width (hw), 640 B rows
constexpr int TP_ = 165;              // LDS row pitch in dwords (160 + 5 pads)

typedef unsigned int  uint4v __attribute__((ext_vector_type(4)));
typedef int           int4v  __attribute__((ext_vector_type(4)));
typedef int           int8v  __attribute__((ext_vector_type(8)));

// ---------------------------------------------------------------------------
// Per-bin bilinear sample table: 16 (pixel-offset, weight) pairs.
// Weight already includes validity mask and the 1/4 sample average.
// ---------------------------------------------------------------------------
struct BilinSamples { int off[16]; float w[16]; };

__device__ __forceinline__ void roi_bin_samples(float x1, float y1,
                                                float hbw, float hbh,
                                                int ph, int pw, int cstride,
                                                BilinSamples& bs) {
    int t = 0;
#pragma unroll
    for (int sy = 0; sy < 2; ++sy) {
        const float y = y1 + ((float)(2 * ph + sy) + 0.5f) * hbh;
#pragma unroll
        for (int sx = 0; sx < 2; ++sx) {
            const float x = x1 + ((float)(2 * pw + sx) + 0.5f) * hbw;
            const bool valid = (y > -1.0f) && (y < (float)H_) &&
                               (x > -1.0f) && (x < (float)W_);
            const float yc = fminf(fmaxf(y, 0.0f), (float)(H_ - 1));
            const float xc = fminf(fmaxf(x, 0.0f), (float)(W_ - 1));
            const int y0 = (int)floorf(yc);
            const int x0 = (int)floorf(xc);
            const int y1i = min(y0 + 1, H_ - 1);
            const int x1i = min(x0 + 1, W_ - 1);
            const float ly = yc - (float)y0, lx = xc - (float)x0;
            const float hy = 1.0f - ly, hx = 1.0f - lx;
            const float vs = valid ? 0.25f : 0.0f;
            bs.off[t] = (y0 * W_ + x0) * cstride;  bs.w[t] = vs * hy * hx; ++t;
            bs.off[t] = (y0 * W_ + x1i) * cstride; bs.w[t] = vs * hy * lx; ++t;
            bs.off[t] = (y1i * W_ + x0) * cstride; bs.w[t] = vs * ly * hx; ++t;
            bs.off[t] = (y1i * W_ + x1i) * cstride; bs.w[t] = vs * ly * lx; ++t;
        }
    }
}

// ---------------------------------------------------------------------------
// Kernel 1 (used): NCHW -> NHWC transpose, tile staged by the Tensor Data
// Mover. One TDM descriptor per block loads a [32 c x 160 hw] tile into LDS
// with hardware-inserted padding (1 dword per 32) -> row pitch 165 dwords,
// making the transposed LDS reads bank-conflict-free. TENSORcnt path.
// ---------------------------------------------------------------------------
__global__ __launch_bounds__(256) void nchw_to_nhwc_tdm(
    const float* __restrict__ src,   // [N, C, HW]
    float* __restrict__ dst)         // [N, HW, C]
{
    __shared__ float s[32 * TP_];            // 21120 B
    const int lane = threadIdx.x & 31;
    const int wave = threadIdx.x >> 5;       // 0..7
    const int hw0  = blockIdx.x * TW_;       // 95 tiles
    const int c0   = blockIdx.y * 32;        // 8 tiles
    const int n    = blockIdx.z;             // 8

    if (wave == 0) {
        const uint64_t ga = (uint64_t)(uintptr_t)(
            src + ((size_t)(n * C_ + c0) * HW_ + hw0));
        const uint32_t lds = (uint32_t)(uintptr_t)&s[0];

        // D# group 0: count=1 | lds_addr | global_addr[56:0] | type=2
        uint4v g0;
        g0[0] = 1u;                                   // count=1, user mode
        g0[1] = lds;                                  // LDS byte address
        g0[2] = (uint32_t)ga;                         // global addr [31:0]
        g0[3] = ((uint32_t)(ga >> 32) & 0x01FFFFFFu)  // global addr [56:32]
                | (2u << 30);                         // type = 2 (image)

        // D# group 1: data_size=4B, pad every 32 dwords by 1 dword,
        // tensor 160x32, tile 160x32, dim0 stride = HW.
        int8v g1;
        g1[0] = (int)((2u << 16)          // data_size = 4 bytes
                      | (1u << 20)        // pad_enable
                      | (4u << 22));      // pad_interval: 32 dwords
                                          // pad_amount = 0 -> 1 dword
        g1[1] = (int)((uint32_t)TW_ << 16);           // tensor_dim0 lo16
        g1[2] = (int)(32u << 16);                     // dim0 hi | tensor_dim1 lo
        g1[3] = (int)((uint32_t)TW_ << 16);           // dim1 hi | tile_dim0
        g1[4] = 32;                                   // tile_dim1 | tile_dim2=0
        g1[5] = HW_;                                  // tensor_dim0_stride lo32
        g1[6] = 0;                                    // stride0 hi | stride1 lo
        g1[7] = 0;                                    // stride1 hi
        int4v gz4 = {0, 0, 0, 0};                     // 2D: groups 2/3 unused
        int8v gz8 = {0, 0, 0, 0, 0, 0, 0, 0};        // unused trailing group

        // clang-23 toolchain: 6-arg form (g0, g1, g2, g3, extra, cpol).
        __builtin_amdgcn_tensor_load_to_lds(g0, g1, gz4, gz4, gz8, 0);
        __builtin_amdgcn_s_wait_tensorcnt(0);
    }
    __syncthreads();

    // Coalesced NHWC writeout: lanes sweep channels (128 B per wave store).
    for (int x = wave; x < TW_; x += 8) {
        dst[(size_t)(n * HW_ + hw0 + x) * C_ + c0 + lane] =
            s[lane * TP_ + x + (x >> 5)];    // stride-165: conflict-free
    }
}

// ---------------------------------------------------------------------------
// Kernel 1b (kept compiled; alternate CDNA5 path): same transpose via the
// per-lane async global->LDS copy (ASYNCcnt). Tile 32x32, LDS pitch 33.
// ---------------------------------------------------------------------------
__global__ __launch_bounds__(256) void nchw_to_nhwc_async(
    const float* __restrict__ src, float* __restrict__ dst)
{
    __shared__ float s[32 * 33];
    const int lane = threadIdx.x & 31;
    const int wave = threadIdx.x >> 5;
    const int hw0  = blockIdx.x * 32;
    const int c0   = blockIdx.y * 32;
    const int n    = blockIdx.z;
#pragma unroll
    for (int j = 0; j < 4; ++j) {
        const int cl = wave * 4 + j;
        const uint64_t ga = (uint64_t)(uintptr_t)(
            src + ((size_t)(n * C_ + c0 + cl) * HW_ + hw0 + lane));
        const uint32_t lds = (uint32_t)(uintptr_t)&s[cl * 33 + lane];
        asm volatile("global_load_async_to_lds_b32 %0, %1, off"
                     :: "v"(lds), "v"(ga) : "memory");
    }
    asm volatile("s_wait_asynccnt 0" ::: "memory");
    __syncthreads();
#pragma unroll
    for (int j = 0; j < 4; ++j) {
        const int hwl = wave * 4 + j;
        dst[(size_t)(n * HW_ + hw0 + hwl) * C_ + c0 + lane] = s[lane * 33 + hwl];
    }
}

// ---------------------------------------------------------------------------
// Kernel 2: RoIAlign gather on NHWC features.
// Block = (roi, channel-half): 8 waves, each wave owns a bin; lanes cover
// 128 channels with one float4 per corner point (coalesced 512 B / point).
// Results go to padded LDS [49][129] so the [C,49] global stores coalesce.
// ---------------------------------------------------------------------------
__global__ __launch_bounds__(256) void roi_gather_nhwc(
    const float* __restrict__ featT,   // [N, HW, C]
    const float* __restrict__ rois,    // [K, 5]
    float* __restrict__ out)           // [K, C, 7, 7]
{
    __shared__ float s_tile[49 * 129];
    const int k     = blockIdx.x;
    const int chunk = blockIdx.y;              // channel half: 0 or 1
    const int lane  = threadIdx.x & 31;
    const int wave  = threadIdx.x >> 5;        // 0..7

    const float* r = rois + (size_t)k * 5;
    const int   b  = (int)r[0];
    const float x1 = r[1] * SCALE_ - 0.5f;
    const float y1 = r[2] * SCALE_ - 0.5f;
    const float x2 = r[3] * SCALE_ - 0.5f;
    const float y2 = r[4] * SCALE_ - 0.5f;
    const float hbw = 0.5f * (x2 - x1) / (float)PW_;
    const float hbh = 0.5f * (y2 - y1) / (float)PH_;

    const float* base = featT + (size_t)b * HW_ * C_ + (size_t)chunk * 128;

    for (int bin = wave; bin < 49; bin += 8) {
        const int ph = bin / 7, pw = bin - ph * 7;
        BilinSamples bs;
        roi_bin_samples(x1, y1, hbw, hbh, ph, pw, C_, bs);

        float4 acc = make_float4(0.f, 0.f, 0.f, 0.f);
#pragma unroll
        for (int p = 0; p < 16; ++p) {
            const float4 v =
                *(reinterpret_cast<const float4*>(base + bs.off[p]) + lane);
            const float w = bs.w[p];
            acc.x = fmaf(w, v.x, acc.x);
            acc.y = fmaf(w, v.y, acc.y);
            acc.z = fmaf(w, v.z, acc.z);
            acc.w = fmaf(w, v.w, acc.w);
        }
        float* d = &s_tile[bin * 129 + lane * 4];
        d[0] = acc.x; d[1] = acc.y; d[2] = acc.z; d[3] = acc.w;
    }
    __syncthreads();

    // Writeout in output order: out[k][chunk*128 + c][pq], pq innermost.
    float* outk = out + (size_t)k * (C_ * 49) + (size_t)chunk * 128 * 49;
    for (int i = threadIdx.x; i < 128 * 49; i += 256) {
        const int cl = i / 49;
        const int pq = i - cl * 49;
        outk[i] = s_tile[pq * 129 + cl];   // stride-129: conflict-free
    }
}

// ---------------------------------------------------------------------------
// Fallback: direct NCHW gather (only if workspace can't hold NHWC copy).
// ---------------------------------------------------------------------------
__global__ __launch_bounds__(256) void roi_direct_nchw(
    const float* __restrict__ feat, const float* __restrict__ rois,
    float* __restrict__ out, int total)
{
    const int idx = blockIdx.x * 256 + threadIdx.x;
    if (idx >= total) return;
    const int pq = idx % 49;
    const int c  = (idx / 49) % C_;
    const int k  = idx / (49 * C_);
    const int ph = pq / 7, pw = pq - ph * 7;

    const float* r = rois + (size_t)k * 5;
    const int   b  = (int)r[0];
    const float x1 = r[1] * SCALE_ - 0.5f;
    const float y1 = r[2] * SCALE_ - 0.5f;
    const float x2 = r[3] * SCALE_ - 0.5f;
    const float y2 = r[4] * SCALE_ - 0.5f;
    const float hbw = 0.5f * (x2 - x1) / (float)PW_;
    const float hbh = 0.5f * (y2 - y1) / (float)PH_;

    BilinSamples bs;
    roi_bin_samples(x1, y1, hbw, hbh, ph, pw, 1, bs);

    const float* plane = feat + ((size_t)b * C_ + c) * HW_;
    float acc = 0.f;
#pragma unroll
    for (int p = 0; p < 16; ++p)
        acc = fmaf(bs.w[p], plane[bs.off[p]], acc);
    out[idx] = acc;
}

// ---------------------------------------------------------------------------
extern "C" void kernel_launch(void* const* d_in, const int* in_sizes, int n_in,
                              void* d_out, int out_size, void* d_ws, size_t ws_size,
                              hipStream_t stream) {
    const float* feat = (const float*)d_in[0];
    const float* rois = (const float*)d_in[1];
    float* out = (float*)d_out;
    const int K = in_sizes[1] / 5;               // 512

    const size_t need = (size_t)N_ * C_ * HW_ * sizeof(float); // 124.5 MB
    if (ws_size >= need) {
        float* featT = (float*)d_ws;
        dim3 g1(HW_ / TW_, C_ / 32, N_);         // 95 x 8 x 8 TDM tiles
        nchw_to_nhwc_tdm<<<g1, 256, 0, stream>>>(feat, featT);
        dim3 g2(K, 2, 1);                        // (roi, channel-half)
        roi_gather_nhwc<<<g2, 256, 0, stream>>>(featT, rois, out);
    } else {
        const int total = K * C_ * 49;
        roi_direct_nchw<<<(total + 255) / 256, 256, 0, stream>>>(feat, rois, out,
                                                                 total);
    }
}